// Maskrcnn_24395414241616
// MI455X (gfx1250) — compile-verified
//
#include <hip/hip_runtime.h>
#include <hip/hip_bf16.h>

#define NB 1000
#define CROPPIX 196
#define CFEAT 256

typedef unsigned int u32;
typedef unsigned int u32x4 __attribute__((ext_vector_type(4)));
typedef unsigned int u32x8 __attribute__((ext_vector_type(8)));
typedef float f2v __attribute__((ext_vector_type(2)));
typedef float f4v __attribute__((ext_vector_type(4)));

// ---------------------------------------------------------------------------
// Kernel B: single-workgroup NMS + top-100 + det assembly.
// Uses TDM (tensor_load_to_lds) to stage refined boxes into LDS, a ballot-
// built 1000x1000 suppression bit-matrix in LDS (~128KB of the 320KB/WGP),
// and a single-wave serial greedy scan over 32-bit alive masks (shfl-based,
// barrier-free).
// ---------------------------------------------------------------------------
__global__ __launch_bounds__(1024) void nms_det_kernel(
    const float* __restrict__ refined,
    const float* __restrict__ scores,
    const u32*   __restrict__ valid,
    const float* __restrict__ coss,
    const float* __restrict__ sins,
    float* __restrict__ det)
{
    __shared__ float sBox[NB*4];        // raw refined boxes (TDM-staged)
    __shared__ float sS[NB];            // raw scores by box index
    __shared__ int   sOrder[NB];        // sorted order (desc score, ties by idx)
    __shared__ float sx1[NB], sy1a[NB], sx2a[NB], sy2a[NB], sA[NB]; // sorted boxes
    __shared__ u32   sValid[32];        // alive-init bits in sorted coords
    __shared__ u32   sup[NB*32];        // suppression bit matrix, row k -> word j
    __shared__ u32   keptW[32];
    __shared__ int   list_[100];
    __shared__ int   nKept;

    int tid  = threadIdx.x;
    int lane = tid & 31;
    int wv   = tid >> 5;

    // ---- TDM: DMA refined boxes (16 KB) into LDS; wave 0 issues once ----
    if (tid < 32) {
        unsigned long long ga = (unsigned long long)(const void*)refined;
        u32 ldsOff = (u32)(unsigned long long)(void*)&sBox[0];
        // D# group 0: count=1, lds_addr, global_addr(57b), type=2
        u32x4 g0 = { 1u,
                     ldsOff,
                     (u32)ga,
                     (u32)((ga >> 32) & 0x1FFFFFFu) | (2u << 30) };
        // D# group 1: wg_mask=0, data_size=4B, tensor_dim0=4000, tensor_dim1=1,
        //             tile_dim0=4000, tile_dim1=1, stride0=4000
        const u32 td0 = 4000u, tile0 = 4000u;
        u32x8 g1 = { (2u << 16),
                     (td0 & 0xFFFFu) << 16,
                     (td0 >> 16) | (1u << 16),
                     (tile0 << 16),
                     1u,
                     4000u,
                     0u,
                     0u };
        u32x4 g2 = { 0u, 0u, 0u, 0u };
        u32x4 g3 = { 0u, 0u, 0u, 0u };
        asm volatile("tensor_load_to_lds %0, %1, %2, %3"
                     :: "s"(g0), "s"(g1), "s"(g2), "s"(g3) : "memory");
        __builtin_amdgcn_s_wait_tensorcnt(0);
    }
    for (int i = tid; i < NB; i += 1024) sS[i] = scores[i];
    __syncthreads();

    // ---- rank sort: argsort(-scores), stable ----
    if (tid < NB) {
        float si = sS[tid];
        int r = 0;
        for (int j = 0; j < NB; ++j) {
            float sj = sS[j];                 // broadcast LDS read
            r += (sj > si) || (sj == si && j < tid);
        }
        sOrder[r] = tid;
    }
    __syncthreads();

    // ---- gather boxes into sorted order; init alive bits ----
    {
        u32 pred = 0;
        if (tid < NB) {
            int b = sOrder[tid];
            float a1 = sBox[b*4+0], a2 = sBox[b*4+1];
            float a3 = sBox[b*4+2], a4 = sBox[b*4+3];
            sx1[tid]=a1; sy1a[tid]=a2; sx2a[tid]=a3; sy2a[tid]=a4;
            sA[tid] = (a3-a1)*(a4-a2);
            pred = valid[b];
        }
        unsigned long long m = __ballot(pred != 0);
        if (lane == 0) sValid[wv] = (u32)m;
    }
    __syncthreads();

    // ---- suppression matrix: 1M IoUs, ballot per 32-col word ----
    for (int cb = 0; cb < 8; ++cb) {
        float cx1[4], cy1[4], cx2[4], cy2[4], ca[4];
        #pragma unroll
        for (int q = 0; q < 4; ++q) {
            int j = cb*128 + q*32 + lane;
            bool in = (j < NB);
            cx1[q] = in ? sx1[j]  : 0.f;
            cy1[q] = in ? sy1a[j] : 0.f;
            cx2[q] = in ? sx2a[j] : 0.f;
            cy2[q] = in ? sy2a[j] : 0.f;
            ca[q]  = in ? sA[j]   : 0.f;
        }
        for (int k = wv; k < NB; k += 32) {
            float rx1 = sx1[k], ry1 = sy1a[k];    // wave-uniform broadcast
            float rx2 = sx2a[k], ry2 = sy2a[k], ra = sA[k];
            #pragma unroll
            for (int q = 0; q < 4; ++q) {
                float xx1 = fmaxf(rx1, cx1[q]);
                float yy1 = fmaxf(ry1, cy1[q]);
                float xx2 = fminf(rx2, cx2[q]);
                float yy2 = fminf(ry2, cy2[q]);
                float inter = fmaxf(xx2-xx1, 0.f) * fmaxf(yy2-yy1, 0.f);
                float iou = inter / (ra + ca[q] - inter + 1e-12f);
                unsigned long long m = __ballot(iou > 0.5f);
                if (lane == 0) sup[k*32 + cb*4 + q] = (u32)m;
            }
        }
    }
    __syncthreads();

    // ---- serial greedy scan, one wave, pure register/LDS bit ops ----
    if (tid < 32) {
        u32 alive = sValid[lane];
        u32 kept  = 0;
        for (int k = 0; k < NB; ++k) {
            int word = k >> 5, bit = k & 31;
            u32 aw = (u32)__shfl((int)alive, word);
            if ((aw >> bit) & 1u) {               // wave-uniform branch
                alive &= ~sup[k*32 + lane];
                if (lane == word) kept |= (1u << bit);
            }
        }
        keptW[lane] = kept;
    }
    __syncthreads();

    // ---- first 100 kept in sorted order ----
    if (tid == 0) {
        int cnt = 0;
        for (int k = 0; k < NB && cnt < 100; ++k)
            if ((keptW[k >> 5] >> (k & 31)) & 1u) list_[cnt++] = sOrder[k];
        nKept = cnt;
    }
    __syncthreads();

    // ---- det rows (zero past nKept, matching * ok) ----
    if (tid < 100) {
        int r = tid;
        if (r < nKept) {
            int idx = list_[r];
            det[r*8+0] = sBox[idx*4+0];
            det[r*8+1] = sBox[idx*4+1];
            det[r*8+2] = sBox[idx*4+2];
            det[r*8+3] = sBox[idx*4+3];
            det[r*8+4] = 1.0f;
            det[r*8+5] = sS[idx];
            det[r*8+6] = coss[idx];
            det[r*8+7] = sins[idx];
        } else {
            #pragma unroll
            for (int t = 0; t < 8; ++t) det[r*8+t] = 0.0f;
        }
    }
}

// ---------------------------------------------------------------------------
// Kernel A: decode_bbox_target + clip_boxes + scores + valid mask
// ---------------------------------------------------------------------------
__global__ void decode_kernel(const float* __restrict__ proposals,
                              const float* __restrict__ box_logits,
                              const float* __restrict__ label_logits,
                              const float* __restrict__ img_shape,
                              float* __restrict__ refined,
                              float* __restrict__ scoresOut,
                              u32* __restrict__ validOut)
{
    int i = blockIdx.x * blockDim.x + threadIdx.x;
    if (i >= NB) return;
    float tx = box_logits[i*4+0] * (1.0f/10.0f);
    float ty = box_logits[i*4+1] * (1.0f/10.0f);
    float tw = box_logits[i*4+2] * (1.0f/5.0f);
    float th = box_logits[i*4+3] * (1.0f/5.0f);
    float ax1 = proposals[i*4+0], ay1 = proposals[i*4+1];
    float ax2 = proposals[i*4+2], ay2 = proposals[i*4+3];
    float wa = ax2-ax1, ha = ay2-ay1;
    float xa = (ax1+ax2)*0.5f, ya = (ay1+ay2)*0.5f;
    const float clipv = 4.6051702f;                 // log(1600/16)
    float wb = expf(fminf(tw, clipv)) * wa;
    float hb = expf(fminf(th, clipv)) * ha;
    float xb = tx*wa + xa;
    float yb = ty*ha + ya;
    float W0 = img_shape[0], W1 = img_shape[1];
    float nx1 = fminf(fmaxf(xb - wb*0.5f, 0.f), W0);
    float ny1 = fminf(fmaxf(yb - hb*0.5f, 0.f), W1);
    float nx2 = fminf(fmaxf(xb + wb*0.5f, 0.f), W0);
    float ny2 = fminf(fmaxf(yb + hb*0.5f, 0.f), W1);
    refined[i*4+0]=nx1; refined[i*4+1]=ny1; refined[i*4+2]=nx2; refined[i*4+3]=ny2;
    float sc = label_logits[i*2+1];
    scoresOut[i] = sc;
    float hh = ny2-ny1, ww = nx2-nx1;
    validOut[i] = (sc > 0.05f && fmaxf(hh*ww, 0.f) > 0.f) ? 1u : 0u;
}

// ---------------------------------------------------------------------------
// Kernel C: multilevel ROI align. Block = one box; thread = 4 consecutive
// crop pixels (49 active threads, 2 waves). Coordinates/weights computed once
// and amortized over the 256-channel loop. X-samples fused into one 8-byte
// gather per row; output written as 16B-aligned non-temporal b128 stores
// (196 floats per channel plane = 49 aligned float4 groups), keeping the
// 200MB output stream out of L2 so the feature pyramid (~85MB) stays resident.
// ---------------------------------------------------------------------------
__global__ __launch_bounds__(64) void roi_align_kernel(
    const float* __restrict__ refined,
    const float* __restrict__ f2, const float* __restrict__ f3,
    const float* __restrict__ f4, const float* __restrict__ f5,
    float* __restrict__ out)
{
    int b = blockIdx.x;
    int t = threadIdx.x;
    if (t >= 49) return;

    float x1 = refined[b*4+0], y1 = refined[b*4+1];
    float x2 = refined[b*4+2], y2 = refined[b*4+3];
    float hh = y2 - y1, ww = x2 - x1;
    float sa = sqrtf(fmaxf(hh*ww, 0.f));
    int lvl = (int)(4.0f + log2f(sa * (1.0f/224.0f) + 1e-6f));  // trunc toward 0
    lvl = min(max(lvl, 2), 5) - 2;

    const float strides[4] = {4.f, 8.f, 16.f, 32.f};
    const int   HS[4]      = {256, 128, 64, 32};
    const float* fptr = (lvl == 0) ? f2 : (lvl == 1) ? f3 : (lvl == 2) ? f4 : f5;
    int H = HS[lvl], W = H;
    float Hf = (float)(H + 2), Wf = (float)(W + 2);   // edge-padded dims
    float inv = 1.0f / strides[lvl];

    float bx1 = x1*inv + 1.f, by1 = y1*inv + 1.f;
    float bx2 = x2*inv + 1.f, by2 = y2*inv + 1.f;
    float stepx = (bx2 - bx1) / 14.0f;
    float stepy = (by2 - by1) / 14.0f;

    float w0A[4], w0B[4], w1A[4], w1B[4];
    int   o0[4],  o1[4];
    #pragma unroll
    for (int u = 0; u < 4; ++u) {
        int pix = t*4 + u;
        int py = pix / 14, px = pix % 14;
        float gx = bx1 - 0.5f + ((float)px + 0.5f) * stepx;
        float gy = by1 - 0.5f + ((float)py + 0.5f) * stepy;
        float ok = (gy > -1.f && gy < Hf && gx > -1.f && gx < Wf) ? 1.f : 0.f;
        float yc = fminf(fmaxf(gy, 0.f), Hf - 1.f);
        float xc = fminf(fmaxf(gx, 0.f), Wf - 1.f);
        int y0 = (int)floorf(yc), x0 = (int)floorf(xc);
        float ly = yc - (float)y0, lx = xc - (float)x0;
        // map padded indices -> unpadded (edge replication == index clamp)
        int ys0 = min(max(y0 - 1, 0), H - 1);
        int ys1 = min(y0, H - 1);
        int xs0 = min(max(x0 - 1, 0), W - 1);
        int xs1 = min(x0, W - 1);
        // x samples always fall in {bxc, bxc+1}: one b64 gather per row,
        // edge duplication folded into the weights
        int bxc = min(xs0, W - 2);
        float wl = 1.f - lx, wr = lx;
        float wxA = ((xs0 == bxc)   ? wl : 0.f) + ((xs1 == bxc)   ? wr : 0.f);
        float wxB = ((xs0 == bxc+1) ? wl : 0.f) + ((xs1 == bxc+1) ? wr : 0.f);
        w0A[u] = (1.f - ly) * wxA * ok;
        w0B[u] = (1.f - ly) * wxB * ok;
        w1A[u] = ly * wxA * ok;
        w1B[u] = ly * wxB * ok;
        o0[u] = ys0 * W + bxc;
        o1[u] = ys1 * W + bxc;
    }

    int HW = H * W;
    const float* p = fptr;
    float* op = out + 800 + (size_t)b * (CFEAT * CROPPIX) + t*4;
    #pragma unroll 2
    for (int c = 0; c < CFEAT; ++c) {
        f4v v;
        #pragma unroll
        for (int u = 0; u < 4; ++u) {
            f2v a, bb;
            __builtin_memcpy(&a,  p + o0[u], 8);   // unaligned-safe 8B gather
            __builtin_memcpy(&bb, p + o1[u], 8);
            v[u] = a.x*w0A[u] + a.y*w0B[u] + bb.x*w1A[u] + bb.y*w1B[u];
        }
        __builtin_nontemporal_store(v, (f4v*)op);  // 16B-aligned NT stream
        p  += HW;
        op += CROPPIX;
    }
}

// ---------------------------------------------------------------------------
extern "C" void kernel_launch(void* const* d_in, const int* in_sizes, int n_in,
                              void* d_out, int out_size, void* d_ws, size_t ws_size,
                              hipStream_t stream) {
    (void)in_sizes; (void)n_in; (void)out_size; (void)ws_size;
    const float* proposals    = (const float*)d_in[0];
    const float* box_logits   = (const float*)d_in[1];
    const float* label_logits = (const float*)d_in[2];
    const float* box_cos      = (const float*)d_in[3];
    const float* box_sin      = (const float*)d_in[4];
    const float* feat2        = (const float*)d_in[5];
    const float* feat3        = (const float*)d_in[6];
    const float* feat4        = (const float*)d_in[7];
    const float* feat5        = (const float*)d_in[8];
    const float* img_shape    = (const float*)d_in[9];

    float* out     = (float*)d_out;
    float* ws      = (float*)d_ws;
    float* refined = ws;                    // 4000 floats
    float* scoresW = ws + 4000;             // 1000 floats
    u32*   validW  = (u32*)(ws + 5000);     // 1000 u32

    decode_kernel<<<4, 256, 0, stream>>>(proposals, box_logits, label_logits,
                                         img_shape, refined, scoresW, validW);
    roi_align_kernel<<<NB, 64, 0, stream>>>(refined, feat2, feat3, feat4, feat5, out);
    nms_det_kernel<<<1, 1024, 0, stream>>>(refined, scoresW, validW,
                                           box_cos, box_sin, out);
}